// RotaryPositionalEmbedding_72395968741920
// MI455X (gfx1250) — compile-verified
//
#include <hip/hip_runtime.h>
#include <hip/hip_bf16.h>
#include <stdint.h>

// RoPE, fp32, B*H*S tokens of D_K=128. Bandwidth-bound (~540 MB @ 23.3 TB/s
// => ~23 us floor). One wave32 per token; lane l owns the float4 holding
// interleaved pairs (2l, 2l+1). x is streamed through LDS via CDNA5 async
// global->LDS DMA (ASYNCcnt) with a 2-deep double buffer per wave; output is
// written with non-temporal b128 stores. sin/cos gathers stay resident in L2
// (4 MB tables, sorted positions). Token index is forced wave-uniform so the
// int64 position fetch becomes one scalar s_load_b64 per token instead of 32
// redundant vector loads.

#define WAVES_PER_BLOCK 8
#define TPW 8  // tokens per wave

typedef __attribute__((ext_vector_type(4))) float fvec4;
typedef __attribute__((ext_vector_type(2))) float fvec2;

__global__ __launch_bounds__(256) void rope_async_kernel(
    const float* __restrict__ x,
    const float* __restrict__ sinT,
    const float* __restrict__ cosT,
    const long long* __restrict__ tpos,
    float* __restrict__ out,
    int n_tokens)
{
    __shared__ float smem[WAVES_PER_BLOCK][2][128];

    const int tid  = threadIdx.x;
    // threadIdx.x >> 5 is wave-uniform under wave32; make that fact explicit
    // so tok0 / g / tpos addressing become SGPR-only (scalar loads, s_cmp).
    const int wave = __builtin_amdgcn_readfirstlane(tid >> 5);
    const int lane = tid & 31;
    const int tok0 = ((int)blockIdx.x * WAVES_PER_BLOCK + wave) * TPW;

    // Flat pointers to __shared__ live in the LDS aperture; low 32 bits are
    // the workgroup-relative LDS byte offset expected by the async-DMA VDST VGPR.
    const uint32_t lds_buf[2] = {
        (uint32_t)(uintptr_t)(&smem[wave][0][lane * 4]),
        (uint32_t)(uintptr_t)(&smem[wave][1][lane * 4])
    };

    // GVS-mode async load: saddr = x base (SGPR pair), per-lane 32-bit byte
    // offset in a VGPR, 16 bytes per lane -> 512 B (one token) per instruction.
    auto issue_async = [&](int g, uint32_t lds_addr) {
        uint32_t voff = (uint32_t)g * 512u + (uint32_t)(lane * 16);
        asm volatile("global_load_async_to_lds_b128 %0, %1, %2"
                     :
                     : "v"(lds_addr), "v"(voff), "s"(x)
                     : "memory");
    };

    // Prologue: prefetch token 0 of this wave's chunk.
    if (tok0 < n_tokens) issue_async(tok0, lds_buf[0]);

    for (int i = 0; i < TPW; ++i) {
        const int g = tok0 + i;           // wave-uniform (SGPR)
        if (g >= n_tokens) break;         // uniform branch -> s_cbranch

        const bool have_next = (i + 1 < TPW) && (g + 1 < n_tokens);
        if (have_next) issue_async(g + 1, lds_buf[(i + 1) & 1]);

        // Async loads complete in order: <=1 outstanding means token i landed.
        if (have_next)
            asm volatile("s_wait_asynccnt 1" ::: "memory");
        else
            asm volatile("s_wait_asynccnt 0" ::: "memory");

        const fvec4 xv = *(const fvec4*)(&smem[wave][i & 1][lane * 4]);

        // Uniform address -> single scalar s_load_b64 (KMcnt), broadcast to
        // the wave for free. Positions < 8192 fit in the low dword.
        const int pos = (int)tpos[g];

        const fvec2 sv = *(const fvec2*)(sinT + (size_t)pos * 64 + lane * 2);
        const fvec2 cv = *(const fvec2*)(cosT + (size_t)pos * 64 + lane * 2);

        fvec4 o;
        o.x = cv.x * xv.x - sv.x * xv.y;   // pair 2*lane
        o.y = sv.x * xv.x + cv.x * xv.y;
        o.z = cv.y * xv.z - sv.y * xv.w;   // pair 2*lane + 1
        o.w = sv.y * xv.z + cv.y * xv.w;

        // Streamed exactly once -> non-temporal b128 store.
        __builtin_nontemporal_store(o, (fvec4*)(out + (size_t)g * 128 + lane * 4));
    }
}

extern "C" void kernel_launch(void* const* d_in, const int* in_sizes, int n_in,
                              void* d_out, int out_size, void* d_ws, size_t ws_size,
                              hipStream_t stream) {
    const float*     x    = (const float*)d_in[0];
    const float*     sinT = (const float*)d_in[1];
    const float*     cosT = (const float*)d_in[2];
    const long long* tpos = (const long long*)d_in[3];
    float*           out  = (float*)d_out;

    const int n_tokens = in_sizes[0] / 128;                 // B*H*S
    const int tokens_per_block = WAVES_PER_BLOCK * TPW;     // 64
    const int grid = (n_tokens + tokens_per_block - 1) / tokens_per_block;

    rope_async_kernel<<<grid, 256, 0, stream>>>(x, sinT, cosT, tpos, out, n_tokens);
}